// GSRAlternative_51908974739615
// MI455X (gfx1250) — compile-verified
//
#include <hip/hip_runtime.h>
#include <hip/hip_bf16.h>

typedef float v2f __attribute__((ext_vector_type(2)));
typedef float v8f __attribute__((ext_vector_type(8)));

#define NEG_SLOPE 0.2f
#define LN_EPS 1e-5f

// ---------- order-preserving float <-> u32 encoding for atomic max ----------
__device__ __forceinline__ unsigned fenc(float f) {
  unsigned u = __float_as_uint(f);
  return (u & 0x80000000u) ? ~u : (u | 0x80000000u);
}
__device__ __forceinline__ float fdec(unsigned u) {
  return (u & 0x80000000u) ? __uint_as_float(u & 0x7fffffffu) : __uint_as_float(~u);
}

// ---------- fills ----------
__global__ void __launch_bounds__(256) fill_f32(float* p, float v, int n) {
  int i = blockIdx.x * 256 + threadIdx.x;
  if (i < n) p[i] = v;
}
__global__ void __launch_bounds__(256) fill_u32(unsigned* p, unsigned v, int n) {
  int i = blockIdx.x * 256 + threadIdx.x;
  if (i < n) p[i] = v;
}

// ---------- fp32 WMMA GEMM: Out[n,128] = X[n,128] @ W[128,128]^T ----------
// Block: 256 thr = 8 waves. Block -> 16 output rows, wave w -> cols [16w,16w+16).
// A (16x4 f32): lanes 0-15 row m, v0=K0 v1=K1; lanes 16-31 row m, v0=K2 v1=K3.
// B (4x16 f32): lane n in 0-15 v0=B[K0,n] v1=B[K1,n]; lanes 16-31 K2/K3.
// C/D (16x16 f32): vgpr r: lanes 0-15 -> M=r, lanes 16-31 -> M=r+8; N=lane&15.
__global__ void __launch_bounds__(256) gemm128_wmma(const float* __restrict__ X,
                                                    const float* __restrict__ Wm,
                                                    float* __restrict__ Out,
                                                    int nrows) {
  constexpr int K = 128;
  constexpr int PAD = 132;  // bank-conflict-free stride for the A access pattern
  __shared__ float sx[16 * PAD];
  const int tid = threadIdx.x;
  const int rowBase = blockIdx.x * 16;
  // cooperative 16x128 tile load: thread t -> row t/16, 8 contiguous floats
  {
    int r = tid >> 4;
    int c = (tid & 15) * 8;
    int rr = rowBase + r;
    if (rr > nrows - 1) rr = nrows - 1;
    const float4* g = (const float4*)(X + (size_t)rr * K + c);
    float4 v0 = g[0], v1 = g[1];
    *(float4*)(sx + r * PAD + c) = v0;
    *(float4*)(sx + r * PAD + c + 4) = v1;
  }
  __syncthreads();
  const int lane = tid & 31;
  const int half = lane >> 4;
  const int mr = lane & 15;
  const int colBase = (tid >> 5) * 16;
  const int bcol = colBase + mr;                 // W row feeding B column
  const v2f* wv = (const v2f*)(Wm + (size_t)bcol * K);
  const float* arow = sx + mr * PAD + 2 * half;
  v8f acc = {0.f, 0.f, 0.f, 0.f, 0.f, 0.f, 0.f, 0.f};
#pragma unroll
  for (int k = 0; k < K; k += 4) {
    v2f a = *(const v2f*)(arow + k);
    v2f b = wv[(k >> 1) + half];
    acc = __builtin_amdgcn_wmma_f32_16x16x4_f32(false, a, false, b, (short)0,
                                                acc, false, false);
  }
#pragma unroll
  for (int r = 0; r < 8; ++r) {
    int orow = rowBase + r + 8 * half;
    if (orow < nrows) Out[(size_t)orow * 128 + colBase + mr] = acc[r];
  }
}

// ---------- per-node attention scores, heads=4 ch=32 ----------
__global__ void __launch_bounds__(256) gat_scores4(const float* __restrict__ h,
                                                   const float* __restrict__ asrc,
                                                   const float* __restrict__ adst,
                                                   float* __restrict__ als,
                                                   float* __restrict__ ald, int n) {
  int t = blockIdx.x * 256 + threadIdx.x;
  int node = t >> 2, head = t & 3;
  if (node >= n) return;
  const float4* hp = (const float4*)(h + (size_t)node * 128 + head * 32);
  const float4* sp = (const float4*)(asrc + head * 32);
  const float4* dp = (const float4*)(adst + head * 32);
  float ss = 0.f, dd = 0.f;
#pragma unroll
  for (int i = 0; i < 8; ++i) {
    float4 hv = hp[i], sv = sp[i], dv = dp[i];
    ss += hv.x * sv.x + hv.y * sv.y + hv.z * sv.z + hv.w * sv.w;
    dd += hv.x * dv.x + hv.y * dv.y + hv.z * dv.z + hv.w * dv.w;
  }
  als[(size_t)node * 4 + head] = ss;
  ald[(size_t)node * 4 + head] = dd;
}

// ---------- edge pass A: leaky-relu score + segment max (encoded atomicMax) ----------
template <int HEADS>
__global__ void __launch_bounds__(256) edge_scores_max(
    const float* __restrict__ als, const float* __restrict__ ald,
    const int* __restrict__ srcA, const int* __restrict__ dstA,
    float* __restrict__ esc, unsigned* __restrict__ menc, int E, int Etot) {
  int e = blockIdx.x * 256 + threadIdx.x;
  if (e >= Etot) return;
  int s, d;
  if (e < E) { s = srcA[e]; d = dstA[e]; } else { s = e - E; d = s; }
#pragma unroll
  for (int h = 0; h < HEADS; ++h) {
    float v = als[(size_t)s * HEADS + h] + ald[(size_t)d * HEADS + h];
    v = (v > 0.f) ? v : NEG_SLOPE * v;
    esc[(size_t)e * HEADS + h] = v;
    atomicMax(menc + (size_t)d * HEADS + h, fenc(v));
  }
}

// ---------- edge pass B: exp(score - max) + segment sum ----------
template <int HEADS>
__global__ void __launch_bounds__(256) edge_exp_sum(
    const int* __restrict__ dstA, float* __restrict__ esc,
    const unsigned* __restrict__ menc, float* __restrict__ denom, int E, int Etot) {
  int e = blockIdx.x * 256 + threadIdx.x;
  if (e >= Etot) return;
  int d = (e < E) ? dstA[e] : (e - E);
#pragma unroll
  for (int h = 0; h < HEADS; ++h) {
    float m = fdec(menc[(size_t)d * HEADS + h]);
    float ex = expf(esc[(size_t)e * HEADS + h] - m);
    esc[(size_t)e * HEADS + h] = ex;
    atomicAdd(denom + (size_t)d * HEADS + h, ex);
  }
}

// ---------- edge pass C: alpha-weighted scatter, 128 channels, wave per edge ----------
__global__ void __launch_bounds__(256) edge_aggr128(
    const float* __restrict__ h, const float* __restrict__ ex,
    const float* __restrict__ denom, const int* __restrict__ srcA,
    const int* __restrict__ dstA, float* __restrict__ out, int E, int Etot) {
  int gt = blockIdx.x * 256 + threadIdx.x;
  int e = gt >> 5;
  if (e >= Etot) return;
  int lane = threadIdx.x & 31;
  int s, d;
  if (e < E) { s = srcA[e]; d = dstA[e]; } else { s = e - E; d = s; }
  int c = lane * 4;
  int head = lane >> 3;  // 32 channels per head
  float alpha = ex[(size_t)e * 4 + head] / denom[(size_t)d * 4 + head];
  float4 hv = *(const float4*)(h + (size_t)s * 128 + c);
  float* op = out + (size_t)d * 128 + c;
  atomicAdd(op + 0, hv.x * alpha);
  atomicAdd(op + 1, hv.y * alpha);
  atomicAdd(op + 2, hv.z * alpha);
  atomicAdd(op + 3, hv.w * alpha);
}

// ---------- edge pass C for layer 5 (heads=1, ch=1) ----------
__global__ void __launch_bounds__(256) edge_aggr1(
    const float* __restrict__ h5, const float* __restrict__ ex,
    const float* __restrict__ denom, const int* __restrict__ srcA,
    const int* __restrict__ dstA, float* __restrict__ out, int E, int Etot) {
  int e = blockIdx.x * 256 + threadIdx.x;
  if (e >= Etot) return;
  int s, d;
  if (e < E) { s = srcA[e]; d = dstA[e]; } else { s = e - E; d = s; }
  float alpha = ex[e] / denom[d];
  atomicAdd(out + d, h5[s] * alpha);
}

// ---------- bias + LayerNorm + ReLU, wave per node ----------
__global__ void __launch_bounds__(256) bias_ln_relu(
    const float* __restrict__ agg, const float* __restrict__ bias,
    const float* __restrict__ g, const float* __restrict__ beta,
    float* __restrict__ outx, int n) {
  int gt = blockIdx.x * 256 + threadIdx.x;
  int node = gt >> 5;
  if (node >= n) return;
  int lane = threadIdx.x & 31;
  float4 v = *(const float4*)(agg + (size_t)node * 128 + lane * 4);
  float4 bb = *(const float4*)(bias + lane * 4);
  v.x += bb.x; v.y += bb.y; v.z += bb.z; v.w += bb.w;
  float s = v.x + v.y + v.z + v.w;
#pragma unroll
  for (int off = 16; off >= 1; off >>= 1) s += __shfl_xor(s, off, 32);
  float mean = s * (1.0f / 128.0f);
  float dx = v.x - mean, dy = v.y - mean, dz = v.z - mean, dw = v.w - mean;
  float sq = dx * dx + dy * dy + dz * dz + dw * dw;
#pragma unroll
  for (int off = 16; off >= 1; off >>= 1) sq += __shfl_xor(sq, off, 32);
  float rs = rsqrtf(sq * (1.0f / 128.0f) + LN_EPS);
  float4 gg = *(const float4*)(g + lane * 4);
  float4 be = *(const float4*)(beta + lane * 4);
  float4 o;
  o.x = fmaxf(dx * rs * gg.x + be.x, 0.f);
  o.y = fmaxf(dy * rs * gg.y + be.y, 0.f);
  o.z = fmaxf(dz * rs * gg.z + be.z, 0.f);
  o.w = fmaxf(dw * rs * gg.w + be.w, 0.f);
  *(float4*)(outx + (size_t)node * 128 + lane * 4) = o;
}

// ---------- layer-5 matvec + scalar scores, wave per node ----------
__global__ void __launch_bounds__(256) matvec_scores5(
    const float* __restrict__ x, const float* __restrict__ W5,
    const float* __restrict__ as5, const float* __restrict__ ad5,
    float* __restrict__ h5, float* __restrict__ als, float* __restrict__ ald, int n) {
  int gt = blockIdx.x * 256 + threadIdx.x;
  int node = gt >> 5;
  if (node >= n) return;
  int lane = threadIdx.x & 31;
  float4 xv = *(const float4*)(x + (size_t)node * 128 + lane * 4);
  float4 wv = *(const float4*)(W5 + lane * 4);
  float s = xv.x * wv.x + xv.y * wv.y + xv.z * wv.z + xv.w * wv.w;
#pragma unroll
  for (int off = 16; off >= 1; off >>= 1) s += __shfl_xor(s, off, 32);
  if (lane == 0) {
    h5[node] = s;
    als[node] = s * as5[0];
    ald[node] = s * ad5[0];
  }
}

__global__ void __launch_bounds__(256) finalize_nodes(const float* __restrict__ agg,
                                                      const float* __restrict__ b5,
                                                      float* __restrict__ outp, int n) {
  int i = blockIdx.x * 256 + threadIdx.x;
  if (i < n) outp[i] = agg[i] + b5[0];
}

__global__ void __launch_bounds__(256) mean_reduce(const float* __restrict__ p,
                                                   float* __restrict__ o, int n) {
  __shared__ float sm[256];
  float s = 0.f;
  for (int i = threadIdx.x; i < n; i += 256) s += p[i];
  sm[threadIdx.x] = s;
  __syncthreads();
  for (int off = 128; off >= 1; off >>= 1) {
    if ((int)threadIdx.x < off) sm[threadIdx.x] += sm[threadIdx.x + off];
    __syncthreads();
  }
  if (threadIdx.x == 0) o[0] = sm[0] / (float)n;
}

extern "C" void kernel_launch(void* const* d_in, const int* in_sizes, int n_in,
                              void* d_out, int out_size, void* d_ws, size_t ws_size,
                              hipStream_t stream) {
  (void)n_in; (void)out_size; (void)ws_size;
  const int IN = 128, H = 128;
  const int N = in_sizes[0] / IN;
  const int E = in_sizes[1] / 2;
  const int Etot = E + N;
  const float* x0 = (const float*)d_in[0];
  const int* ei = (const int*)d_in[1];
  const int* srcA = ei;
  const int* dstA = ei + E;
  auto up = [&](int i) { return (const float*)d_in[i]; };

  // workspace carve-up (256B aligned)
  char* wp = (char*)d_ws;
  auto carve = [&](size_t bytes) {
    void* p = wp;
    wp += (bytes + 255) & ~(size_t)255;
    return p;
  };
  float* hbuf = (float*)carve((size_t)N * H * 4);
  float* aggbuf = (float*)carve((size_t)N * H * 4);
  float* xbuf = (float*)carve((size_t)N * H * 4);
  float* als = (float*)carve((size_t)N * 4 * 4);
  float* ald = (float*)carve((size_t)N * 4 * 4);
  unsigned* menc = (unsigned*)carve((size_t)N * 4 * 4);
  float* denom = (float*)carve((size_t)N * 4 * 4);
  float* esc = (float*)carve((size_t)Etot * 4 * 4);

  dim3 blk(256);
  auto cdiv = [](long a, long b) { return (unsigned)((a + b - 1) / b); };

  const float* cur = x0;
  for (int L = 1; L <= 4; ++L) {
    const float* Wm = up(3 + (L - 1) * 4);
    const float* as = up(4 + (L - 1) * 4);
    const float* ad = up(5 + (L - 1) * 4);
    const float* bb = up(6 + (L - 1) * 4);
    const float* gg = up(23 + (L - 1) * 2);
    const float* be = up(24 + (L - 1) * 2);
    fill_f32<<<cdiv((long)N * H, 256), blk, 0, stream>>>(aggbuf, 0.f, N * H);
    fill_u32<<<cdiv((long)N * 4, 256), blk, 0, stream>>>(menc, 0u, N * 4);
    fill_f32<<<cdiv((long)N * 4, 256), blk, 0, stream>>>(denom, 0.f, N * 4);
    gemm128_wmma<<<cdiv(N, 16), blk, 0, stream>>>(cur, Wm, hbuf, N);
    gat_scores4<<<cdiv((long)N * 4, 256), blk, 0, stream>>>(hbuf, as, ad, als, ald, N);
    edge_scores_max<4><<<cdiv(Etot, 256), blk, 0, stream>>>(als, ald, srcA, dstA, esc, menc, E, Etot);
    edge_exp_sum<4><<<cdiv(Etot, 256), blk, 0, stream>>>(dstA, esc, menc, denom, E, Etot);
    edge_aggr128<<<cdiv((long)Etot * 32, 256), blk, 0, stream>>>(hbuf, esc, denom, srcA, dstA, aggbuf, E, Etot);
    bias_ln_relu<<<cdiv((long)N * 32, 256), blk, 0, stream>>>(aggbuf, bb, gg, be, xbuf, N);
    cur = xbuf;
  }

  // layer 5: H -> 1, heads=1, concat=False (mean over 1 head == identity)
  float* out5 = aggbuf;
  fill_f32<<<cdiv(N, 256), blk, 0, stream>>>(out5, 0.f, N);
  fill_u32<<<cdiv(N, 256), blk, 0, stream>>>(menc, 0u, N);
  fill_f32<<<cdiv(N, 256), blk, 0, stream>>>(denom, 0.f, N);
  matvec_scores5<<<cdiv((long)N * 32, 256), blk, 0, stream>>>(cur, up(19), up(20), up(21), hbuf, als, ald, N);
  edge_scores_max<1><<<cdiv(Etot, 256), blk, 0, stream>>>(als, ald, srcA, dstA, esc, menc, E, Etot);
  edge_exp_sum<1><<<cdiv(Etot, 256), blk, 0, stream>>>(dstA, esc, menc, denom, E, Etot);
  edge_aggr1<<<cdiv(Etot, 256), blk, 0, stream>>>(hbuf, esc, denom, srcA, dstA, out5, E, Etot);

  float* outp = (float*)d_out;
  finalize_nodes<<<cdiv(N, 256), blk, 0, stream>>>(out5, up(22), outp, N);
  mean_reduce<<<1, blk, 0, stream>>>(outp, outp + N, N);
}